// RecurrentDrafting_65721589563917
// MI455X (gfx1250) — compile-verified
//
#include <hip/hip_runtime.h>
#include <hip/hip_bf16.h>
#include <float.h>

// Problem constants (from the reference).
#define B_    32
#define W_    8
#define L_    6
#define LM1_  5
#define V_    32000
#define H_    4096
#define TILE_ 1024        // 256 threads * 4 floats per tile of a vocab row
#define SEG_  4           // vocab segments per batch row (occupancy)
#define SEGLEN_ 8192      // ceil-ish segment length (last segment shorter)

// ---------------------------------------------------------------------------
// CDNA5 async global<->LDS DMA path (gfx1250), inline asm (portable across
// ROCm 7.2 / amdgpu-toolchain; verified to assemble in round 1).
// ---------------------------------------------------------------------------
__device__ __forceinline__ void async_load_f128(unsigned lds_byte_off, const float* gaddr) {
  asm volatile("global_load_async_to_lds_b128 %0, %1, off"
               :: "v"(lds_byte_off), "v"(gaddr) : "memory");
}
__device__ __forceinline__ void async_store_f128(float* gaddr, unsigned lds_byte_off) {
  asm volatile("global_store_async_from_lds_b128 %0, %1, off"
               :: "v"(gaddr), "v"(lds_byte_off) : "memory");
}
template <int N>
__device__ __forceinline__ void wait_async_le() {
  asm volatile("s_wait_asynccnt %0" :: "i"(N) : "memory");
}

// ---------------------------------------------------------------------------
// Deterministic counter-based RNG (Threefry-2x32, 20 rounds) -> 2 uniforms.
// ---------------------------------------------------------------------------
__device__ __forceinline__ unsigned rotl32(unsigned x, unsigned r) {
  return (x << r) | (x >> (32u - r));
}
__device__ __forceinline__ void threefry2x32(unsigned k0, unsigned k1,
                                             unsigned c0, unsigned c1,
                                             unsigned& o0, unsigned& o1) {
  unsigned ks0 = k0, ks1 = k1, ks2 = 0x1BD11BDAu ^ k0 ^ k1;
  unsigned x0 = c0 + ks0, x1 = c1 + ks1;
#define TF_R4(a, b, c, d)                          \
  x0 += x1; x1 = rotl32(x1, a); x1 ^= x0;          \
  x0 += x1; x1 = rotl32(x1, b); x1 ^= x0;          \
  x0 += x1; x1 = rotl32(x1, c); x1 ^= x0;          \
  x0 += x1; x1 = rotl32(x1, d); x1 ^= x0;
  TF_R4(13, 15, 26, 6)  x0 += ks1; x1 += ks2 + 1u;
  TF_R4(17, 29, 16, 24) x0 += ks2; x1 += ks0 + 2u;
  TF_R4(13, 15, 26, 6)  x0 += ks0; x1 += ks1 + 3u;
  TF_R4(17, 29, 16, 24) x0 += ks1; x1 += ks2 + 4u;
  TF_R4(13, 15, 26, 6)  x0 += ks2; x1 += ks0 + 5u;
#undef TF_R4
  o0 = x0; o1 = x1;
}
__device__ __forceinline__ float u01(unsigned bits) {
  return (float)(bits >> 8) * (1.0f / 16777216.0f);   // [0,1)
}

// Order-preserving float->uint map (larger score => larger key).
__device__ __forceinline__ unsigned mono_key(float f) {
  const unsigned u = __float_as_uint(f);
  return (u & 0x80000000u) ? ~u : (u | 0x80000000u);
}

// ---------------------------------------------------------------------------
// Kernel 1: acceptance scan + workspace init. One wave32 per batch row b.
// ---------------------------------------------------------------------------
__global__ void rd_accept_kernel(const int* __restrict__ beams,
                                 const float* __restrict__ llm,
                                 const float* __restrict__ drafter,
                                 int* __restrict__ ws_n, int* __restrict__ ws_seq,
                                 unsigned long long* __restrict__ ws_key,
                                 float* __restrict__ out_n, float* __restrict__ out_seq) {
  const int b = blockIdx.x;
  const int lane = threadIdx.x;       // blockDim.x == 32 (one wave)
  int n = 0;
  if (lane < W_) {
    const size_t bw = (size_t)b * W_ + lane;
    int run = 0;
    for (int t = 0; t < LM1_; ++t) {
      const int tok = beams[bw * L_ + t + 1];                        // drafted token
      const float lpd = drafter[(bw * LM1_ + t) * (size_t)V_ + tok];
      const float lpl = llm[(bw * L_ + t) * (size_t)V_ + tok];
      unsigned r0, r1;
      threefry2x32(0x6a09e667u, 42u, (unsigned)(bw * LM1_ + t), 0u, r0, r1);
      run += (u01(r0) < __expf(lpl - lpd)) ? 1 : 0;                  // accept test
      n += (run == t + 1) ? 1 : 0;                                   // prefix length
    }
  }
  // wave32 max/argmax over the 8 beams (first max wins ties, as jnp.argmax).
  int bn = (lane < W_) ? n : -1;
  int bw_ = (lane < W_) ? lane : 0;
  for (int off = 16; off > 0; off >>= 1) {
    const int on = __shfl_down(bn, off);
    const int ow = __shfl_down(bw_, off);
    if (on > bn || (on == bn && ow < bw_)) { bn = on; bw_ = ow; }
  }
  if (lane == 0) {
    ws_n[b] = bn;  ws_seq[b] = bw_;
    ws_key[b] = 0ull;                 // re-init every call (determinism)
    out_n[b] = (float)bn;  out_seq[b] = (float)bw_;
  }
}

// ---------------------------------------------------------------------------
// Kernel 2 (fused): blocks [0, B*SEG)      -> Gumbel-max over a V-segment,
//                   blocks [B*SEG, +B)     -> hidden row copy via async DMA.
// Vocab rows are double-buffered through LDS with the async engine: issue
// stage t+1, s_wait_asynccnt <= (ops in flight for t+1), consume stage t.
// Per-block partial argmaxes merge via global_atomic_max_u64 on a key that
// orders by score then by lower vocab index.
// ---------------------------------------------------------------------------
__global__ void rd_finish_kernel(const float* __restrict__ llm,
                                 const float* __restrict__ drafter,
                                 const float* __restrict__ hidden,
                                 const int* __restrict__ ws_n,
                                 const int* __restrict__ ws_seq,
                                 unsigned long long* __restrict__ ws_key,
                                 float* __restrict__ out_state) {
  // One shared object => LDS offset 0 base for the asm byte offsets.
  // Layout: 2 stages x (rowA tile | rowB tile) = 16 KB, then 16-float red area.
  __shared__ float smem[4 * TILE_ + 16];

  const int tid = threadIdx.x;                    // 256 threads = 8 wave32
  const bool is_copy = (blockIdx.x >= B_ * SEG_);

  if (is_copy) {
    const int b = blockIdx.x - B_ * SEG_;
    const int n = ws_n[b];
    const size_t bw = (size_t)b * W_ + ws_seq[b];
    const float* src = hidden + (bw * L_ + n) * (size_t)H_;
    float* dst = out_state + (size_t)b * H_;
    // 4 tile loads into 4 LDS regions, one wait, 4 async stores.
#pragma unroll
    for (int tt = 0; tt < 4; ++tt)
      async_load_f128((unsigned)(tt * 4096 + tid * 16), src + tt * TILE_ + tid * 4);
    wait_async_le<0>();                           // all LDS writes landed
#pragma unroll
    for (int tt = 0; tt < 4; ++tt)
      async_store_f128(dst + tt * TILE_ + tid * 4, (unsigned)(tt * 4096 + tid * 16));
    wait_async_le<0>();                           // stores drained
    return;
  }

  // ---- sampling over one V-segment ----
  const int b   = blockIdx.x / SEG_;
  const int seg = blockIdx.x % SEG_;
  const int segstart = seg * SEGLEN_;
  const int seglen = min(SEGLEN_, V_ - segstart);
  const int n = ws_n[b];
  const size_t bw = (size_t)b * W_ + ws_seq[b];
  const bool entire = (n == LM1_);
  const int ni = n - (entire ? 1 : 0);
  const float* pA = (entire ? (llm + (bw * L_ + (L_ - 1)) * (size_t)V_)
                            : (llm + (bw * L_ + ni) * (size_t)V_)) + segstart;
  const float* pB = (drafter + (bw * LM1_ + ni) * (size_t)V_) + segstart;

  const int ntiles = (seglen + TILE_ - 1) / TILE_;  // <= 8
  const int wavebase = (tid & ~31) * 4;             // wave's first element in a tile

  // Issue one tile's async loads into stage buffer (stage in {0,1}).
  auto issue_tile = [&](int tt, int stage) {
    const int base = tt * TILE_ + tid * 4;
    if (base < seglen) {                            // V%4==0 -> full b128 ok
      const unsigned lo = (unsigned)(stage * (2 * TILE_ * 4) + tid * 16);
      async_load_f128(lo, pA + base);
      if (!entire) async_load_f128(lo + TILE_ * 4, pB + base);
    }
  };

  float best = -__builtin_inff();
  int besti = 0;

  issue_tile(0, 0);
  for (int t = 0; t < ntiles; ++t) {
    if (t + 1 < ntiles) issue_tile(t + 1, (t + 1) & 1);
    // ops in flight for stage t+1 for *this wave* (issuance is per-wave).
    int k = 0;
    if (t + 1 < ntiles && ((t + 1) * TILE_ + wavebase) < seglen) k = entire ? 1 : 2;
    if (k == 0)      wait_async_le<0>();
    else if (k == 1) wait_async_le<1>();
    else             wait_async_le<2>();
    // consume stage t (each lane reads exactly the 16B it loaded).
    const int base = t * TILE_ + tid * 4;
    if (base < seglen) {
      const int soff = (t & 1) * (2 * TILE_);
      unsigned r0 = 0, r1 = 0;
#pragma unroll
      for (int j = 0; j < 4; ++j) {
        const int v = segstart + base + j;
        const float a = smem[soff + tid * 4 + j];
        float p;
        if (entire) {
          p = __expf(a);
        } else {
          const float d = smem[soff + TILE_ + tid * 4 + j];
          p = fmaxf(__expf(a) - __expf(d), 0.0f);
        }
        const float logit = __logf(fmaxf(p, FLT_MIN));
        if ((j & 1) == 0)  // one Threefry call feeds two elements
          threefry2x32(0xbb67ae85u, 42u, (unsigned)b, (unsigned)(v >> 1), r0, r1);
        const float u = fmaxf(u01((j & 1) ? r1 : r0), 1.0e-30f);
        const float g = -__logf(-__logf(u));        // Gumbel(0,1)
        const float sc = logit + g;
        if (sc > best || (sc == best && v < besti)) { best = sc; besti = v; }
      }
    }
  }

  // wave32 argmax reduction
  for (int off = 16; off > 0; off >>= 1) {
    const float ob = __shfl_down(best, off);
    const int oi = __shfl_down(besti, off);
    if (ob > best || (ob == best && oi < besti)) { best = ob; besti = oi; }
  }
  const int wave = tid >> 5;
  if ((tid & 31) == 0) {
    smem[4 * TILE_ + wave] = best;
    smem[4 * TILE_ + 8 + wave] = __int_as_float(besti);
  }
  __syncthreads();
  if (tid == 0) {
    float bb = smem[4 * TILE_];
    int bi = __float_as_int(smem[4 * TILE_ + 8]);
    for (int wv = 1; wv < 8; ++wv) {
      const float ob = smem[4 * TILE_ + wv];
      const int oi = __float_as_int(smem[4 * TILE_ + 8 + wv]);
      if (ob > bb || (ob == bb && oi < bi)) { bb = ob; bi = oi; }
    }
    // key: [ mono(score) | (V - idx) ]  -> atomicMax picks best score,
    // lower index on ties. Deterministic regardless of arrival order.
    const unsigned long long key =
        ((unsigned long long)mono_key(bb) << 32) | (unsigned)(V_ - bi);
    atomicMax(&ws_key[b], key);                    // global_atomic_max_u64
  }
}

// ---------------------------------------------------------------------------
// Kernel 3: decode winning keys -> next_tokens (one wave total).
// ---------------------------------------------------------------------------
__global__ void rd_decode_kernel(const unsigned long long* __restrict__ ws_key,
                                 float* __restrict__ out_tok) {
  const int b = threadIdx.x;                       // 32 lanes == B
  if (b < B_) {
    const unsigned long long key = ws_key[b];
    out_tok[b] = (float)(V_ - (int)(unsigned)(key & 0xffffffffull));
  }
}

// ---------------------------------------------------------------------------
// Host launcher. Inputs (setup_inputs order): beams (int), log_probs_by_llm,
// log_probs_by_drafter, last_hidden_state (all f32). Output layout (flat,
// return order): state[B*H], next_tokens[B], n_tokens_in_seq[B], seq_in_beam[B].
// ---------------------------------------------------------------------------
extern "C" void kernel_launch(void* const* d_in, const int* in_sizes, int n_in,
                              void* d_out, int out_size, void* d_ws, size_t ws_size,
                              hipStream_t stream) {
  const int*   beams   = (const int*)d_in[0];
  const float* llm     = (const float*)d_in[1];
  const float* drafter = (const float*)d_in[2];
  const float* hidden  = (const float*)d_in[3];

  float* out       = (float*)d_out;
  float* out_state = out;                         // B*H
  float* out_tok   = out + (size_t)B_ * H_;       // B
  float* out_n     = out_tok + B_;                // B
  float* out_seq   = out_n + B_;                  // B

  int* ws_n   = (int*)d_ws;                                   // B ints
  int* ws_seq = ws_n + B_;                                    // B ints
  unsigned long long* ws_key =
      (unsigned long long*)((char*)d_ws + 256);               // B u64 (8-aligned)

  rd_accept_kernel<<<B_, 32, 0, stream>>>(beams, llm, drafter,
                                          ws_n, ws_seq, ws_key, out_n, out_seq);
  rd_finish_kernel<<<B_ * SEG_ + B_, 256, 0, stream>>>(llm, drafter, hidden,
                                                       ws_n, ws_seq, ws_key, out_state);
  rd_decode_kernel<<<1, 32, 0, stream>>>(ws_key, out_tok);
}